// TextureFeatureExtractor_51625506898362
// MI455X (gfx1250) — compile-verified
//
#include <hip/hip_runtime.h>
#include <math.h>

typedef __attribute__((ext_vector_type(2))) float v2f;
typedef __attribute__((ext_vector_type(8))) float v8f;

#define LVL 256
#define HW 512
#define NTHREADS 1024

// U weight matrix: 256 x 16, col0 = 1, col1 = i, col2 = i^2, rest 0.
// Branchless: lowers to v_cndmask + adds (no EXEC-mask divergence inside the
// WMMA loop, which previously cost ~50 extra SALU/branch ops per iteration).
__device__ __forceinline__ float Uval(int i, int c) {
    float iv = (float)i;
    float r  = (c == 0) ? 1.0f : 0.0f;
    r += (c == 1) ? iv : 0.0f;
    r += (c == 2) ? iv * iv : 0.0f;
    return r;
}

__global__ __launch_bounds__(NTHREADS) void glcm_feat_kernel(
        const int* __restrict__ x, float* __restrict__ feat) {
    extern __shared__ unsigned int hist[];   // 65536 bins = 256 KB (CDNA5 320KB LDS)
    __shared__ float  yscr[16 * 256];        // per-wave Y (16x16) re-layout bounce
    __shared__ float  mscr[16 * 256];        // per-wave partial M (16x16)
    __shared__ double red[32 * 3];           // wave partials: energy, dissim, homog

    const int tid  = threadIdx.x;
    const int lane = tid & 31;
    const int wv   = tid >> 5;
    const int imgIdx = blockIdx.x >> 2;
    const int angle  = blockIdx.x & 3;
    const int* __restrict__ img = x + (size_t)imgIdx * HW * HW;

    // ---- Phase A: zero histogram ----
    for (int b = tid; b < LVL * LVL; b += NTHREADS) hist[b] = 0u;
    __syncthreads();

    // ---- Phase B: build GLCM with LDS atomics (ds_add_u32) ----
    int rmax, cmax, aor, aoc, bor, boc;
    switch (angle) {
        case 0:  rmax = 512; cmax = 511; aor = 0; aoc = 0; bor = 0; boc = 1; break; // 0
        case 1:  rmax = 511; cmax = 511; aor = 0; aoc = 0; bor = 1; boc = 1; break; // pi/4
        case 2:  rmax = 511; cmax = 512; aor = 0; aoc = 0; bor = 1; boc = 0; break; // pi/2
        default: rmax = 511; cmax = 511; aor = 0; aoc = 1; bor = 1; boc = 0; break; // 3pi/4
    }
    const int npairs = rmax * cmax;
    for (int p = tid; p < npairs; p += NTHREADS) {
        int r, c;
        if (cmax == 512) { r = p >> 9; c = p & 511; }
        else             { r = p / 511; c = p - r * 511; }
        int a = img[(r + aor) * HW + (c + aoc)];
        int b = img[(r + bor) * HW + (c + boc)];
        atomicAdd(&hist[a * LVL + b], 1u);
    }
    __syncthreads();

    // ---- Phase C: non-separable per-bin sums over Q = h + h^T ----
    double e_sum = 0.0, d_sum = 0.0, h_sum = 0.0;
    for (int bin = tid; bin < LVL * LVL; bin += NTHREADS) {
        int i = bin >> 8, j = bin & 255;
        double q = (double)(hist[bin] + hist[j * LVL + i]);
        int d  = i - j;
        int ad = (d < 0) ? -d : d;
        e_sum += q * q;                           // -> energy numerator
        d_sum += q * (double)ad;                  // -> dissimilarity numerator
        h_sum += q / (1.0 + (double)(d * d));     // -> homogeneity numerator
    }
    for (int off = 16; off > 0; off >>= 1) {      // wave32 reduction
        e_sum += __shfl_down(e_sum, off, 32);
        d_sum += __shfl_down(d_sum, off, 32);
        h_sum += __shfl_down(h_sum, off, 32);
    }
    if (lane == 0) {
        red[wv * 3 + 0] = e_sum;
        red[wv * 3 + 1] = d_sum;
        red[wv * 3 + 2] = h_sum;
    }

    // ---- Phase D step 1: Y = Q * U via v_wmma_f32_16x16x4_f32 ----
    // Wave bi (0..15) owns row-block bi. f32 A-layout (16x4 in 2 VGPRs):
    // lanes0-15 hold K={0,1}, lanes16-31 hold K={2,3}; M = lane%16. B mirrored.
    if (wv < 16) {
        const int bi    = wv;
        const int m     = lane & 15;
        const int khalf = (lane < 16) ? 0 : 2;
        const int rowA  = bi * 16 + m;
        v8f Yacc = {0.f, 0.f, 0.f, 0.f, 0.f, 0.f, 0.f, 0.f};
        for (int bj = 0; bj < 16; ++bj) {
#pragma unroll
            for (int kk = 0; kk < 16; kk += 4) {
                int col0 = bj * 16 + kk + khalf;
                v2f a, b;
                // Q_ij = h_ij + h_ji, symmetrized on the fly (exact in f32)
                a.x = (float)(hist[rowA * LVL + col0]     + hist[col0 * LVL + rowA]);
                a.y = (float)(hist[rowA * LVL + col0 + 1] + hist[(col0 + 1) * LVL + rowA]);
                b.x = Uval(col0, m);
                b.y = Uval(col0 + 1, m);
                Yacc = __builtin_amdgcn_wmma_f32_16x16x4_f32(
                    false, a, false, b, (short)0, Yacc, false, false);
            }
        }
#pragma unroll
        for (int v = 0; v < 8; ++v) {             // D layout -> row-major 16x16
            int row = (lane < 16) ? v : v + 8;
            yscr[bi * 256 + row * 16 + m] = Yacc[v];
        }
    }
    __syncthreads();

    // ---- Phase D step 2: M_partial = U[bi]^T * Y_bi ----
    if (wv < 16) {
        const int bi    = wv;
        const int m     = lane & 15;
        const int khalf = (lane < 16) ? 0 : 2;
        v8f Macc = {0.f, 0.f, 0.f, 0.f, 0.f, 0.f, 0.f, 0.f};
#pragma unroll
        for (int kk = 0; kk < 16; kk += 4) {
            int k0 = kk + khalf;
            v2f a, b;
            a.x = Uval(bi * 16 + k0,     m);       // (U^T)[m][k] = U[16*bi+k][m]
            a.y = Uval(bi * 16 + k0 + 1, m);
            b.x = yscr[bi * 256 + k0 * 16 + m];
            b.y = yscr[bi * 256 + (k0 + 1) * 16 + m];
            Macc = __builtin_amdgcn_wmma_f32_16x16x4_f32(
                false, a, false, b, (short)0, Macc, false, false);
        }
#pragma unroll
        for (int v = 0; v < 8; ++v) {
            int row = (lane < 16) ? v : v + 8;
            mscr[bi * 256 + row * 16 + m] = Macc[v];
        }
    }
    __syncthreads();

    // ---- Final: thread 0 combines M = sum_bi M_partial and per-bin sums ----
    if (tid == 0) {
        double esum = 0.0, dsum = 0.0, hsum = 0.0;
        for (int w = 0; w < 32; ++w) {
            esum += red[w * 3 + 0];
            dsum += red[w * 3 + 1];
            hsum += red[w * 3 + 2];
        }
        float S0 = 0.f, Si = 0.f, Sii = 0.f, Sjj = 0.f, Sij = 0.f;
        for (int w = 0; w < 16; ++w) {
            const float* Mw = &mscr[w * 256];
            S0  += Mw[0 * 16 + 0];   // sum Q            (= 2N)
            Si  += Mw[1 * 16 + 0];   // sum Q * i
            Sii += Mw[2 * 16 + 0];   // sum Q * i^2
            Sjj += Mw[0 * 16 + 2];   // sum Q * j^2
            Sij += Mw[1 * 16 + 1];   // sum Q * i*j
        }
        float inv      = 1.0f / S0;
        float contrast = (Sii + Sjj - 2.0f * Sij) * inv;
        float dissim   = (float)dsum * inv;
        float homog    = (float)hsum * inv;
        float energy   = sqrtf((float)esum) * inv;
        float mu       = Si * inv;                    // mu_i == mu_j (symmetric)
        float var      = Sii * inv - mu * mu;         // std_i * std_j
        float cov      = Sij * inv - mu * mu;
        float corr     = (var < 1e-15f) ? 1.0f : cov / fmaxf(var, 1e-30f);
        float* o = feat + blockIdx.x * 5;
        o[0] = contrast; o[1] = dissim; o[2] = homog; o[3] = energy; o[4] = corr;
    }
}

// Average the 4 angles per image: feat[256*5] -> out[64*5]
__global__ void glcm_avg_kernel(const float* __restrict__ feat, float* __restrict__ out) {
    int t = blockIdx.x * blockDim.x + threadIdx.x;
    if (t >= 64 * 5) return;
    int im = t / 5, f = t % 5;
    float s = 0.f;
#pragma unroll
    for (int a = 0; a < 4; ++a) s += feat[(im * 4 + a) * 5 + f];
    out[t] = 0.25f * s;
}

extern "C" void kernel_launch(void* const* d_in, const int* in_sizes, int n_in,
                              void* d_out, int out_size, void* d_ws, size_t ws_size,
                              hipStream_t stream) {
    (void)in_sizes; (void)n_in; (void)out_size; (void)ws_size;
    const int* x  = (const int*)d_in[0];
    float* out    = (float*)d_out;
    float* feat   = (float*)d_ws;   // 256 * 5 floats of scratch

    // 256 KB dynamic LDS per workgroup (CDNA5 supports up to 320 KB/WGP)
    (void)hipFuncSetAttribute(reinterpret_cast<const void*>(glcm_feat_kernel),
                              hipFuncAttributeMaxDynamicSharedMemorySize,
                              LVL * LVL * (int)sizeof(unsigned int));

    glcm_feat_kernel<<<dim3(64 * 4), dim3(NTHREADS),
                       LVL * LVL * sizeof(unsigned int), stream>>>(x, feat);
    glcm_avg_kernel<<<dim3(1), dim3(320), 0, stream>>>(feat, out);
}